// MambaBlock_87205015978006
// MI455X (gfx1250) — compile-verified
//
#include <hip/hip_runtime.h>
#include <hip/hip_bf16.h>

typedef float v2f __attribute__((ext_vector_type(2)));
typedef float v8f __attribute__((ext_vector_type(8)));
typedef unsigned int u32x4 __attribute__((ext_vector_type(4)));
typedef int i32x4 __attribute__((ext_vector_type(4)));
typedef int i32x8 __attribute__((ext_vector_type(8)));

#define B_SZ   4
#define LSEQ   2048
#define DMODEL 1024
#define DINNER 2048
#define DSTATE 16
#define DCONV  4
#define MROWS  (B_SZ * LSEQ)   // 8192

#if __has_builtin(__builtin_amdgcn_tensor_load_to_lds) && \
    __has_builtin(__builtin_amdgcn_s_wait_tensorcnt)
#define USE_TDM 1
#else
#define USE_TDM 0
#endif

// ---------------------------------------------------------------------------
// LayerNorm: one 256-thread block per row of 1024
// ---------------------------------------------------------------------------
__global__ __launch_bounds__(256) void ln_kernel(const float* __restrict__ x,
                                                 const float* __restrict__ w,
                                                 const float* __restrict__ b,
                                                 float* __restrict__ xn) {
    int row = blockIdx.x;
    int tid = threadIdx.x;
    const float* xr = x + (size_t)row * DMODEL;
    float4 v = *(const float4*)(xr + tid * 4);
    float s  = v.x + v.y + v.z + v.w;
    float sq = v.x * v.x + v.y * v.y + v.z * v.z + v.w * v.w;
    __shared__ float s1[256], s2[256];
    s1[tid] = s; s2[tid] = sq;
    __syncthreads();
    for (int st = 128; st > 0; st >>= 1) {
        if (tid < st) { s1[tid] += s1[tid + st]; s2[tid] += s2[tid + st]; }
        __syncthreads();
    }
    float mean = s1[0] * (1.0f / DMODEL);
    float var  = s2[0] * (1.0f / DMODEL) - mean * mean;
    float rstd = rsqrtf(var + 1e-5f);
    float4 wv = *(const float4*)(w + tid * 4);
    float4 bv = *(const float4*)(b + tid * 4);
    float4 o;
    o.x = (v.x - mean) * rstd * wv.x + bv.x;
    o.y = (v.y - mean) * rstd * wv.y + bv.y;
    o.z = (v.z - mean) * rstd * wv.z + bv.z;
    o.w = (v.w - mean) * rstd * wv.w + bv.w;
    *(float4*)(xn + (size_t)row * DMODEL + tid * 4) = o;
}

// ---------------------------------------------------------------------------
// TDM helper: issue one 2-D tile load (global -> LDS) via the Tensor Data
// Mover.  data_size = 4B.  LDS padding done by TDM pad feature so the LDS
// image matches the padded arrays used by the WMMA fragment reads.
// Toolchain uses the 6-arg builtin: (g0 u32x4, g1 i32x8, g2 i32x4, g3 i32x4,
// extra i32x8, cpol i32); groups 2/3 + extra are zero for 2-D tensors.
// ---------------------------------------------------------------------------
#if USE_TDM
__device__ __forceinline__ void tdm_load_tile_f32(unsigned lds_addr,
                                                  const float* gptr,
                                                  int tile_d0, int tile_d1,
                                                  int stride0_elems,
                                                  int pad_interval_code,
                                                  int pad_amount_code) {
    unsigned long long ga = (unsigned long long)(size_t)gptr;
    u32x4 g0;
    g0[0] = 1u;                                     // count=1, user descriptor
    g0[1] = lds_addr;                               // LDS byte address
    g0[2] = (unsigned)(ga & 0xFFFFFFFFu);           // global_addr[31:0]
    g0[3] = (unsigned)((ga >> 32) & 0x01FFFFFFu)    // global_addr[56:32]
            | (2u << 30);                           // type = 2 (image)
    // Oversized tensor dims so no element is OOB-dropped (tiles are exact).
    const int tdim0 = stride0_elems;                // >= tile_d0
    const int tdim1 = 1 << 20;                      // >= tile_d1
    i32x8 g1;
    g1[0] = (2 << 16)                               // data_size = 4 bytes
            | (1 << 20)                             // pad_enable
            | (pad_interval_code << 22)             // pad every N dwords
            | (pad_amount_code << 25);              // pad amount (code+1 dw)
    g1[1] = (tdim0 & 0xFFFF) << 16;                 // tensor_dim0[15:0]
    g1[2] = ((tdim0 >> 16) & 0xFFFF) | ((tdim1 & 0xFFFF) << 16);
    g1[3] = ((tdim1 >> 16) & 0xFFFF) | (tile_d0 << 16);
    g1[4] = tile_d1 & 0xFFFF;                       // tile_dim1 (tile_dim2=0)
    g1[5] = stride0_elems;                          // tensor_dim0_stride lo
    g1[6] = 0;
    g1[7] = 0;
    i32x4 z4 = {0, 0, 0, 0};
    i32x8 z8 = {0, 0, 0, 0, 0, 0, 0, 0};
    __builtin_amdgcn_tensor_load_to_lds(g0, g1, z4, z4, z8, 0);
}
#endif

// ---------------------------------------------------------------------------
// Generic fp32 WMMA GEMM: C[M,N] = A[M,K] @ W[K,N] with epilogues.
// Block tile 128x128, 8 waves (4x2), wave tile 32x64 (2x4 WMMA frags).
// K staged in 16-wide slabs, double-buffered via the TDM when available.
// EPI 0: split columns at N/2 -> out0 / out1 (W_in -> x_pre, z)
// EPI 1: +bias[col], softplus -> out0       (W_delta -> delta)
// EPI 2: +aux[row*N+col] residual -> out0   (W_out -> final)
// ---------------------------------------------------------------------------
#define BMT 128
#define BNT 128
#define KBT 16

template <int EPI>
__global__ __launch_bounds__(256) void wmma_gemm(const float* __restrict__ A,
                                                 const float* __restrict__ W,
                                                 float* __restrict__ out0,
                                                 float* __restrict__ out1,
                                                 const float* __restrict__ aux,
                                                 int M, int N, int K) {
    // A rows padded 16+4 dwords (TDM pad codes 3/3), B rows 128+4 (codes 6/3)
    __shared__ float As[2][BMT][KBT + 4];   // 2 x 10240 B
    __shared__ float Bs[2][KBT][BNT + 4];   // 2 x  8448 B

    int tid  = threadIdx.x;
    int lane = tid & 31;
    int wid  = tid >> 5;
    int half = lane >> 4;
    int l16  = lane & 15;
    int wm   = wid >> 1;     // 0..3
    int wn   = wid & 1;      // 0..1
    int blockRow = blockIdx.y * BMT;
    int blockCol = blockIdx.x * BNT;

    v8f acc[2][4];
#pragma unroll
    for (int tm = 0; tm < 2; ++tm)
#pragma unroll
        for (int tn = 0; tn < 4; ++tn)
#pragma unroll
            for (int i = 0; i < 8; ++i) acc[tm][tn][i] = 0.0f;

    auto compute_slab = [&](int buf) {
#pragma unroll
        for (int kk = 0; kk < KBT; kk += 4) {
            v2f af[2], bf[4];
#pragma unroll
            for (int tm = 0; tm < 2; ++tm) {
                int r = wm * 32 + tm * 16 + l16;
                af[tm].x = As[buf][r][kk + 2 * half];
                af[tm].y = As[buf][r][kk + 2 * half + 1];
            }
#pragma unroll
            for (int tn = 0; tn < 4; ++tn) {
                int c = wn * 64 + tn * 16 + l16;
                bf[tn].x = Bs[buf][kk + 2 * half][c];
                bf[tn].y = Bs[buf][kk + 2 * half + 1][c];
            }
#pragma unroll
            for (int tm = 0; tm < 2; ++tm)
#pragma unroll
                for (int tn = 0; tn < 4; ++tn)
                    acc[tm][tn] = __builtin_amdgcn_wmma_f32_16x16x4_f32(
                        false, af[tm], false, bf[tn], (short)0, acc[tm][tn],
                        false, false);
        }
    };

#if USE_TDM
    // -------- TDM double-buffered pipeline (wave 0 drives the DMA) --------
    auto issue_slab = [&](int buf, int k0) {
        tdm_load_tile_f32((unsigned)(size_t)&As[buf][0][0],
                          A + (size_t)blockRow * K + k0,
                          /*tile_d0=*/KBT, /*tile_d1=*/BMT,
                          /*stride0=*/K, /*padI(16dw)=*/3, /*padA(4dw)=*/3);
        tdm_load_tile_f32((unsigned)(size_t)&Bs[buf][0][0],
                          W + (size_t)k0 * N + blockCol,
                          /*tile_d0=*/BNT, /*tile_d1=*/KBT,
                          /*stride0=*/N, /*padI(128dw)=*/6, /*padA(4dw)=*/3);
    };
    if (wid == 0) {
        issue_slab(0, 0);
        __builtin_amdgcn_s_wait_tensorcnt(0);
    }
    __syncthreads();
    for (int k0 = 0; k0 < K; k0 += KBT) {
        int cur = (k0 / KBT) & 1;
        if (wid == 0 && (k0 + KBT) < K) issue_slab(cur ^ 1, k0 + KBT);
        compute_slab(cur);                        // overlaps with DMA
        if (wid == 0) __builtin_amdgcn_s_wait_tensorcnt(0);
        __syncthreads();
    }
#else
    // -------- fallback: synchronous staged loads --------
    for (int k0 = 0; k0 < K; k0 += KBT) {
#pragma unroll
        for (int i = 0; i < 2; ++i) {            // A tile 128x16
            int f = tid + i * 256;
            int r = f >> 2;
            int c = (f & 3) * 4;
            *(float4*)&As[0][r][c] =
                *(const float4*)&A[(size_t)(blockRow + r) * K + k0 + c];
        }
#pragma unroll
        for (int i = 0; i < 2; ++i) {            // B tile 16x128
            int f = tid + i * 256;
            int r = f >> 5;
            int c = (f & 31) * 4;
            *(float4*)&Bs[0][r][c] =
                *(const float4*)&W[(size_t)(k0 + r) * N + blockCol + c];
        }
        __syncthreads();
        compute_slab(0);
        __syncthreads();
    }
#endif

    // -------- epilogue --------
#pragma unroll
    for (int tm = 0; tm < 2; ++tm) {
#pragma unroll
        for (int tn = 0; tn < 4; ++tn) {
            int col = blockCol + wn * 64 + tn * 16 + l16;
#pragma unroll
            for (int r = 0; r < 8; ++r) {
                int row = blockRow + wm * 32 + tm * 16 + half * 8 + r;
                float v = acc[tm][tn][r];
                if (EPI == 0) {
                    int hN = N >> 1;
                    if (col < hN)
                        out0[(size_t)row * hN + col] = v;
                    else
                        out1[(size_t)row * hN + (col - hN)] = v;
                } else if (EPI == 1) {
                    v += aux[col];
                    v = (v > 20.0f) ? v : log1pf(__expf(v));
                    out0[(size_t)row * N + col] = v;
                } else {
                    v += aux[(size_t)row * N + col];
                    out0[(size_t)row * N + col] = v;
                }
            }
        }
    }
}

// ---------------------------------------------------------------------------
// Depthwise causal conv (K=4) + SiLU, one thread per (b,t,d)
// ---------------------------------------------------------------------------
__global__ __launch_bounds__(256) void conv_silu_kernel(const float* __restrict__ xpre,
                                                        const float* __restrict__ cw,
                                                        const float* __restrict__ cb,
                                                        float* __restrict__ xpath) {
    size_t i = (size_t)blockIdx.x * 256 + threadIdx.x;   // over MROWS*DINNER
    int d = (int)(i % DINNER);
    size_t bt = i / DINNER;
    int t = (int)(bt % LSEQ);
    float acc = cb[d];
#pragma unroll
    for (int k = 0; k < DCONV; ++k) {
        int ts = t - (DCONV - 1) + k;
        if (ts >= 0)
            acc += cw[d * DCONV + k] * xpre[(bt - (size_t)(DCONV - 1 - k)) * DINNER + d];
    }
    xpath[i] = acc / (1.0f + __expf(-acc));   // silu
}

// ---------------------------------------------------------------------------
// B/C projections: x_path[8192,2048] @ W{b,c}[2048,16] -> [8192,16]
// One wave per (16-row tile, sel); N=16 is exactly one WMMA tile.
// ---------------------------------------------------------------------------
__global__ __launch_bounds__(256) void gemm_bc_kernel(const float* __restrict__ xpath,
                                                      const float* __restrict__ Wb,
                                                      const float* __restrict__ Wc,
                                                      float* __restrict__ Bm,
                                                      float* __restrict__ Cm) {
    int gwid = (blockIdx.x * 256 + threadIdx.x) >> 5;   // 0..1023
    int lane = threadIdx.x & 31;
    int half = lane >> 4, l16 = lane & 15;
    int tile = gwid >> 1;
    int sel  = gwid & 1;
    const float* W = sel ? Wc : Wb;
    float* O = sel ? Cm : Bm;
    int row0 = tile * 16;

    v8f acc;
#pragma unroll
    for (int i = 0; i < 8; ++i) acc[i] = 0.0f;

    for (int k = 0; k < DINNER; k += 4) {
        v2f a, w;
        const float* ap = xpath + (size_t)(row0 + l16) * DINNER + k + 2 * half;
        a.x = ap[0];
        a.y = ap[1];
        const float* wp = W + (size_t)(k + 2 * half) * DSTATE + l16;
        w.x = wp[0];
        w.y = wp[DSTATE];
        acc = __builtin_amdgcn_wmma_f32_16x16x4_f32(false, a, false, w, (short)0,
                                                    acc, false, false);
    }
#pragma unroll
    for (int r = 0; r < 8; ++r)
        O[(size_t)(row0 + half * 8 + r) * DSTATE + l16] = acc[r];
}

// ---------------------------------------------------------------------------
// Selective scan: one lane per (channel, state). 16 lanes = one channel.
// Loop-carried dep is a single FMA; exp/shuffles are off the critical path.
// Fused epilogue: G = (h.C + D*x) * silu(z)
// ---------------------------------------------------------------------------
#define SCHUNK 128
__global__ __launch_bounds__(256) void scan_kernel(const float* __restrict__ xpath,
                                                   const float* __restrict__ delta,
                                                   const float* __restrict__ Bm,
                                                   const float* __restrict__ Cm,
                                                   const float* __restrict__ z,
                                                   const float* __restrict__ A_log,
                                                   const float* __restrict__ Dp_,
                                                   float* __restrict__ G) {
    int b  = blockIdx.x / (DINNER / 16);
    int d0 = (blockIdx.x % (DINNER / 16)) * 16;
    int tid  = threadIdx.x;
    int lane = tid & 31;
    int n    = lane & 15;
    int cw   = lane >> 4;
    int wv   = tid >> 5;
    int d    = d0 + wv * 2 + cw;

    __shared__ float Bs[SCHUNK][DSTATE];
    __shared__ float Cs[SCHUNK][DSTATE];

    float A_n = -__expf(A_log[(size_t)d * DSTATE + n]);
    float Dp  = Dp_[d];
    float h   = 0.0f;
    const size_t baseBL = (size_t)b * LSEQ;

    for (int t0 = 0; t0 < LSEQ; t0 += SCHUNK) {
        __syncthreads();
        for (int i = tid; i < SCHUNK * DSTATE / 4; i += 256) {
            ((float4*)&Bs[0][0])[i] = ((const float4*)(Bm + (baseBL + t0) * DSTATE))[i];
            ((float4*)&Cs[0][0])[i] = ((const float4*)(Cm + (baseBL + t0) * DSTATE))[i];
        }
        __syncthreads();
        for (int tt = 0; tt < SCHUNK; ++tt) {
            size_t idx = (baseBL + t0 + tt) * DINNER + d;
            float xt = xpath[idx];
            float dt = delta[idx];
            float ab = __expf(dt * A_n);
            h = ab * h + (dt * xt) * Bs[tt][n];
            float p = h * Cs[tt][n];
            p += __shfl_xor(p, 8, 16);
            p += __shfl_xor(p, 4, 16);
            p += __shfl_xor(p, 2, 16);
            p += __shfl_xor(p, 1, 16);
            if (n == 0) {
                float zz = z[idx];
                float sz = zz / (1.0f + __expf(-zz));
                G[idx] = (p + Dp * xt) * sz;
            }
        }
    }
}

// ---------------------------------------------------------------------------
// Launch
// ---------------------------------------------------------------------------
extern "C" void kernel_launch(void* const* d_in, const int* in_sizes, int n_in,
                              void* d_out, int out_size, void* d_ws, size_t ws_size,
                              hipStream_t stream) {
    const float* x       = (const float*)d_in[0];
    const float* norm_w  = (const float*)d_in[1];
    const float* norm_b  = (const float*)d_in[2];
    const float* W_in    = (const float*)d_in[3];
    const float* conv_w  = (const float*)d_in[4];
    const float* conv_b  = (const float*)d_in[5];
    const float* A_log   = (const float*)d_in[6];
    const float* W_b     = (const float*)d_in[7];
    const float* W_c     = (const float*)d_in[8];
    const float* W_delta = (const float*)d_in[9];
    const float* b_delta = (const float*)d_in[10];
    const float* D_param = (const float*)d_in[11];
    const float* W_out   = (const float*)d_in[12];
    float* out = (float*)d_out;

    char* ws = (char*)d_ws;
    const size_t SZ_XN    = (size_t)MROWS * DMODEL * 4;   // 32 MB
    const size_t SZ_INNER = (size_t)MROWS * DINNER * 4;   // 64 MB
    const size_t SZ_BC    = (size_t)MROWS * DSTATE * 4;   // 0.5 MB
    size_t off = 0;
    float* xn    = (float*)(ws + off); off += SZ_XN;
    float* xpre  = (float*)(ws + off); off += SZ_INNER;
    float* zbuf  = (float*)(ws + off); off += SZ_INNER;
    float* xpath = (float*)(ws + off); off += SZ_INNER;
    float* dlt   = (float*)(ws + off); off += SZ_INNER;
    float* Bmat  = (float*)(ws + off); off += SZ_BC;
    float* Cmat  = (float*)(ws + off); off += SZ_BC;
    float* G     = (float*)(ws + off); off += SZ_INNER;

    // 1. LayerNorm
    ln_kernel<<<MROWS, 256, 0, stream>>>(x, norm_w, norm_b, xn);

    // 2. xz = xn @ W_in  -> x_pre | z   (M=8192, N=4096, K=1024)
    wmma_gemm<0><<<dim3(4096 / BNT, MROWS / BMT), 256, 0, stream>>>(
        xn, W_in, xpre, zbuf, nullptr, MROWS, 2 * DINNER, DMODEL);

    // 3. depthwise causal conv + SiLU
    conv_silu_kernel<<<(MROWS * DINNER) / 256, 256, 0, stream>>>(
        xpre, conv_w, conv_b, xpath);

    // 4. delta = softplus(x_path @ W_delta + b_delta)  (N=K=2048)
    wmma_gemm<1><<<dim3(DINNER / BNT, MROWS / BMT), 256, 0, stream>>>(
        xpath, W_delta, dlt, nullptr, b_delta, MROWS, DINNER, DINNER);

    // 5. B/C projections (N=16 WMMA tiles)
    gemm_bc_kernel<<<(MROWS / 16) * 2 / 8, 256, 0, stream>>>(
        xpath, W_b, W_c, Bmat, Cmat);

    // 6. selective scan fused with y * silu(z)
    scan_kernel<<<B_SZ * (DINNER / 16), 256, 0, stream>>>(
        xpath, dlt, Bmat, Cmat, zbuf, A_log, D_param, G);

    // 7. out = G @ W_out + residual  (N=1024, K=2048)
    wmma_gemm<2><<<dim3(DMODEL / BNT, MROWS / BMT), 256, 0, stream>>>(
        G, W_out, out, nullptr, x, MROWS, DMODEL, DINNER);
}